// EncoderSPECls_21397527069092
// MI455X (gfx1250) — compile-verified
//
#include <hip/hip_runtime.h>
#include <math.h>

// ---------------------------------------------------------------------------
// EncoderSPECls for MI455X (gfx1250, wave32).
// FP32 pipeline; kNN distance GEMM uses V_WMMA_F32_16X16X4_F32.
// ---------------------------------------------------------------------------

#define BB   16     // batch
#define KNN  24     // neighbors

typedef float v2f __attribute__((ext_vector_type(2)));
typedef float v8f __attribute__((ext_vector_type(8)));

__device__ __forceinline__ float spe_coef(int f, int F) {
  // BETA / ALPHA^(f/F) = 100 * exp(-ln(1000) * f/F)
  return 100.0f * __expf(-6.9077552789821368f * ((float)f / (float)F));
}

// -------------------------- initial SPE embedding --------------------------
// feat0[b,n,j] for out_dim=64: feat_dim=11, feat_num=66, idx=linspace(0,65,64)
__global__ void embed0_kernel(const float* __restrict__ xyz,
                              float* __restrict__ feat, int total) {
  int i = blockIdx.x * blockDim.x + threadIdx.x;
  if (i >= total) return;
  int j  = i & 63;
  int pn = i >> 6;                       // b*N + n
  int idx = (j == 63) ? 65 : (int)((double)j * (65.0 / 63.0));
  int dsel = idx / 22;                   // which coordinate
  int f    = (idx % 22) >> 1;            // frequency index
  float x = xyz[(size_t)pn * 3 + dsel];
  float a = spe_coef(f, 11) * x;
  feat[i] = (idx & 1) ? cosf(a) : sinf(a);
}

// ------------------------------- FPS ---------------------------------------
__global__ void __launch_bounds__(256)
fps_kernel(const float* __restrict__ xyz, int N, int S,
           int* __restrict__ fps_idx, float* __restrict__ xyz_out) {
  __shared__ float dist[2048];
  __shared__ float sval[256];
  __shared__ int   sidx[256];
  int b = blockIdx.x, tid = threadIdx.x;
  const float* p = xyz + (size_t)b * N * 3;
  for (int i = tid; i < N; i += 256) dist[i] = 1e10f;
  __syncthreads();
  int farthest = 0;
  for (int s = 0; s < S; ++s) {
    if (tid == 0) {
      fps_idx[b * S + s] = farthest;
      xyz_out[((size_t)b * S + s) * 3 + 0] = p[farthest * 3 + 0];
      xyz_out[((size_t)b * S + s) * 3 + 1] = p[farthest * 3 + 1];
      xyz_out[((size_t)b * S + s) * 3 + 2] = p[farthest * 3 + 2];
    }
    float cx = p[farthest * 3 + 0];
    float cy = p[farthest * 3 + 1];
    float cz = p[farthest * 3 + 2];
    float bv = -1.0f; int bi = 0x7fffffff;
    for (int i = tid; i < N; i += 256) {
      float dx = p[i * 3 + 0] - cx;
      float dy = p[i * 3 + 1] - cy;
      float dz = p[i * 3 + 2] - cz;
      float d = dx * dx + dy * dy + dz * dz;
      float m = fminf(dist[i], d);
      dist[i] = m;
      if (m > bv) { bv = m; bi = i; }   // per-thread keeps lowest index on ties
    }
    sval[tid] = bv; sidx[tid] = bi;
    __syncthreads();
    for (int off = 128; off > 0; off >>= 1) {
      if (tid < off) {
        float ov = sval[tid + off]; int oi = sidx[tid + off];
        if (ov > sval[tid] || (ov == sval[tid] && oi < sidx[tid])) {
          sval[tid] = ov; sidx[tid] = oi;
        }
      }
      __syncthreads();
    }
    farthest = sidx[0];
    __syncthreads();
  }
}

// ------------------------------- kNN (WMMA) --------------------------------
// Wave per 16-query tile. A = [16x4]: (qx,qy,qz,1). B = [4x16]:
// (-2px,-2py,-2pz,|p|^2)  =>  C[m][n] = |p_n|^2 - 2 q_m . p_n
// (|q|^2 is row-constant; irrelevant for per-row top-k ranking.)
__global__ void __launch_bounds__(32)
knn_kernel(const float* __restrict__ xyz_pts, const float* __restrict__ xyz_q,
           int N, int S, int* __restrict__ knn_out) {
  int b    = blockIdx.y;
  int m0   = blockIdx.x * 16;
  int lane = threadIdx.x;
  int mr   = lane & 15;
  const float* q = xyz_q   + ((size_t)b * S + m0) * 3;
  const float* P = xyz_pts +  (size_t)b * N * 3;

  // A fragment: lanes 0-15 -> K=0,1 ; lanes 16-31 -> K=2,3 (row M = lane&15)
  v2f afrag;
  if (lane < 16) { afrag.x = q[mr * 3 + 0]; afrag.y = q[mr * 3 + 1]; }
  else           { afrag.x = q[mr * 3 + 2]; afrag.y = 1.0f; }

  float bd[KNN]; int bidx[KNN];
#pragma unroll
  for (int t = 0; t < KNN; ++t) { bd[t] = 3.0e38f; bidx[t] = 0; }
  float worstV = 3.0e38f; int worstSlot = 0;

  __shared__ float tile[16][17];

  for (int nt = 0; nt < N; nt += 16) {
    int n = nt + mr;
    float px = P[n * 3 + 0], py = P[n * 3 + 1], pz = P[n * 3 + 2];
    // B fragment: lanes 0-15 -> K=0,1 rows; lanes 16-31 -> K=2,3 rows (col N = lane&15)
    v2f bfrag;
    if (lane < 16) { bfrag.x = -2.0f * px; bfrag.y = -2.0f * py; }
    else           { bfrag.x = -2.0f * pz; bfrag.y = px * px + py * py + pz * pz; }

    v8f c = {};
    c = __builtin_amdgcn_wmma_f32_16x16x4_f32(false, afrag, false, bfrag,
                                              (short)0, c, false, false);
    // C layout: VGPR r holds M = r + 8*(lane/16), N = lane&15
    int mbase = (lane < 16) ? 0 : 8;
#pragma unroll
    for (int r = 0; r < 8; ++r) tile[mbase + r][mr] = c[r];
    __syncthreads();

    if (lane < 16) {
#pragma unroll
      for (int jj = 0; jj < 16; ++jj) {
        float d = tile[lane][jj];
        if (d < worstV) {                       // strict <: keeps earliest on ties
#pragma unroll
          for (int t = 0; t < KNN; ++t)
            if (t == worstSlot) { bd[t] = d; bidx[t] = nt + jj; }
          float wv = -3.0e38f; int wsl = 0;
#pragma unroll
          for (int t = 0; t < KNN; ++t)
            if (bd[t] > wv) { wv = bd[t]; wsl = t; }
          worstV = wv; worstSlot = wsl;
        }
      }
    }
    __syncthreads();
  }
  if (lane < 16) {
    int* o = knn_out + ((size_t)b * S + m0 + lane) * KNN;
#pragma unroll
    for (int t = 0; t < KNN; ++t) o[t] = bidx[t];
  }
}

// -------------------- global std reduction (ddof handled later) ------------
__global__ void center_stats_kernel(const float* __restrict__ xyz_pts,
                                    const float* __restrict__ xyz_q,
                                    const float* __restrict__ feat_in,
                                    const int* __restrict__ fps_idx,
                                    const int* __restrict__ knn_idx,
                                    int N, int S, int C,
                                    float* __restrict__ stats) {
  int gid = blockIdx.x * blockDim.x + threadIdx.x;
  int total = BB * S * KNN;
  float sx = 0.f, sx2 = 0.f, sf = 0.f, sf2 = 0.f;
  for (int t = gid; t < total; t += gridDim.x * blockDim.x) {
    int b = t / (S * KNN);
    int rem = t - b * S * KNN;
    int s = rem / KNN;
    int k = rem - s * KNN;
    int nn = knn_idx[((size_t)b * S + s) * KNN + k];
    const float* pq = xyz_q + ((size_t)b * S + s) * 3;
    const float* pp = xyz_pts + ((size_t)b * N + nn) * 3;
    for (int d = 0; d < 3; ++d) { float v = pp[d] - pq[d]; sx += v; sx2 += v * v; }
    int fi = fps_idx[b * S + s];
    const float* fq = feat_in + ((size_t)b * N + fi) * C;
    const float* fp = feat_in + ((size_t)b * N + nn) * C;
    for (int c = 0; c < C; ++c) { float v = fp[c] - fq[c]; sf += v; sf2 += v * v; }
  }
  for (int off = 16; off > 0; off >>= 1) {
    sx  += __shfl_down(sx,  off);
    sx2 += __shfl_down(sx2, off);
    sf  += __shfl_down(sf,  off);
    sf2 += __shfl_down(sf2, off);
  }
  if ((threadIdx.x & 31) == 0) {
    atomicAdd(&stats[0], sx);  atomicAdd(&stats[1], sx2);
    atomicAdd(&stats[2], sf);  atomicAdd(&stats[3], sf2);
  }
}

// ----------- fused SPE + weighted max/mean pooling + BN-stat accum ---------
__global__ void __launch_bounds__(256)
stage_main_kernel(const float* __restrict__ xyz_pts, const float* __restrict__ xyz_q,
                  const float* __restrict__ feat_in, const int* __restrict__ fps_idx,
                  const int* __restrict__ knn_idx, const float* __restrict__ stats,
                  int N, int S, int C, int Cout, int F, int feat_num,
                  float* __restrict__ feat_out, float* __restrict__ bnstats) {
  int s = blockIdx.x, b = blockIdx.y, tid = threadIdx.x;
  __shared__ float sxyz[KNN][3];
  __shared__ int   snn[KNN];
  __shared__ float sinv[2];
  if (tid == 0) {
    double n3 = (double)BB * (double)S * (double)KNN * 3.0;
    double v3 = ((double)stats[1] - (double)stats[0] * (double)stats[0] / n3) / (n3 - 1.0);
    if (v3 < 0.0) v3 = 0.0;
    sinv[0] = 1.0f / ((float)sqrt(v3) + 1e-5f);
    double nf = (double)BB * (double)S * (double)KNN * (double)C;
    double vf = ((double)stats[3] - (double)stats[2] * (double)stats[2] / nf) / (nf - 1.0);
    if (vf < 0.0) vf = 0.0;
    sinv[1] = 1.0f / ((float)sqrt(vf) + 1e-5f);
  }
  __syncthreads();
  if (tid < KNN) {
    int nn = knn_idx[((size_t)b * S + s) * KNN + tid];
    snn[tid] = nn;
    float inv = sinv[0];
    for (int d = 0; d < 3; ++d)
      sxyz[tid][d] = (xyz_pts[((size_t)b * N + nn) * 3 + d] -
                      xyz_q[((size_t)b * S + s) * 3 + d]) * inv;
  }
  __syncthreads();
  int fi = fps_idx[b * S + s];
  float invf = sinv[1];
  double step = (double)(feat_num - 1) / (double)(Cout - 1);
  for (int j = tid; j < Cout; j += 256) {
    int idx = (j == Cout - 1) ? (feat_num - 1) : (int)(step * (double)j);
    if (idx > feat_num - 1) idx = feat_num - 1;
    int dsel = idx / (2 * F);
    int f    = (idx % (2 * F)) >> 1;
    int par  = idx & 1;
    float coef = spe_coef(f, F);
    int cj = (j < C) ? j : (j - C);
    float fsj = feat_in[((size_t)b * N + fi) * C + cj];
    float wmax = -3.0e38f, wsum = 0.0f;
    for (int k = 0; k < KNN; ++k) {
      float arg = coef * sxyz[k][dsel];
      float pe  = par ? cosf(arg) : sinf(arg);
      float fv  = (j < C) ? (feat_in[((size_t)b * N + snn[k]) * C + j] - fsj) * invf
                          : fsj;   // concat half: raw feat_s broadcast
      float w = (fv + pe) * pe;
      wmax = fmaxf(wmax, w);
      wsum += w;
    }
    float a = wmax + wsum * (1.0f / (float)KNN);
    feat_out[((size_t)b * S + s) * Cout + j] = a;
    atomicAdd(&bnstats[j], a);
    atomicAdd(&bnstats[Cout + j], a * a);
  }
}

// --------------------------- BatchNorm + exact GELU ------------------------
__global__ void bn_gelu_kernel(float* __restrict__ feat,
                               const float* __restrict__ bnstats,
                               const float* __restrict__ gamma,
                               const float* __restrict__ beta,
                               int S, int Cout) {
  int total = BB * S * Cout;
  int i = blockIdx.x * blockDim.x + threadIdx.x;
  if (i >= total) return;
  int j = i % Cout;
  float cnt  = (float)(BB * S);
  float mean = bnstats[j] / cnt;
  float var  = bnstats[Cout + j] / cnt - mean * mean;
  float x = feat[i];
  float y = gamma[j] * (x - mean) * rsqrtf(var + 1e-5f) + beta[j];
  feat[i] = 0.5f * y * (1.0f + erff(y * 0.70710678118654752f));
}

// ------------------------------- head --------------------------------------
__global__ void final_kernel(const float* __restrict__ feat,
                             float* __restrict__ out) {
  int j = blockIdx.x * blockDim.x + threadIdx.x;  // 0..1023
  int b = blockIdx.y;
  float mx = -3.0e38f, sm = 0.0f;
  for (int s = 0; s < 128; ++s) {
    float v = feat[((size_t)b * 128 + s) * 1024 + j];
    mx = fmaxf(mx, v); sm += v;
  }
  out[(size_t)b * 1024 + j] = mx + sm * (1.0f / 128.0f);
}

__global__ void zero_kernel(float* __restrict__ p, int n) {
  int i = blockIdx.x * blockDim.x + threadIdx.x;
  if (i < n) p[i] = 0.0f;
}

// ---------------------------------------------------------------------------
extern "C" void kernel_launch(void* const* d_in, const int* in_sizes, int n_in,
                              void* d_out, int out_size, void* d_ws, size_t ws_size,
                              hipStream_t stream) {
  (void)in_sizes; (void)n_in; (void)out_size; (void)ws_size;
  const float* xyz_in0 = (const float*)d_in[0];
  const float* gammas[4] = {(const float*)d_in[1], (const float*)d_in[3],
                            (const float*)d_in[5], (const float*)d_in[7]};
  const float* betas[4]  = {(const float*)d_in[2], (const float*)d_in[4],
                            (const float*)d_in[6], (const float*)d_in[8]};

  char* ws = (char*)d_ws;
  size_t o = 0;
  float* xyzA   = (float*)(ws + o); o += (size_t)16 * 1024 * 3 * 4;
  float* xyzB   = (float*)(ws + o); o += (size_t)16 * 1024 * 3 * 4;
  float* featA  = (float*)(ws + o); o += (size_t)16 * 2048 * 64 * 4;   // 8 MB
  float* featB  = (float*)(ws + o); o += (size_t)16 * 2048 * 64 * 4;   // 8 MB
  int*   fpsidx = (int*)  (ws + o); o += (size_t)16 * 1024 * 4;
  int*   knnidx = (int*)  (ws + o); o += (size_t)16 * 1024 * KNN * 4;
  float* stats  = (float*)(ws + o); o += 256;
  float* bnst   = (float*)(ws + o); o += (size_t)2 * 1024 * 4;

  // initial feature embedding: [16,2048,64]
  {
    int total = 16 * 2048 * 64;
    embed0_kernel<<<(total + 255) / 256, 256, 0, stream>>>(xyz_in0, featA, total);
  }

  const int Fs[4]  = {22, 43, 86, 171};     // feat_dim per stage
  const int FNs[4] = {132, 258, 516, 1026}; // feat_num per stage

  const float* xyz_cur = xyz_in0;
  float* feat_cur = featA;
  int N = 2048, C = 64;

  for (int i = 0; i < 4; ++i) {
    int S = N >> 1;
    int Cout = C << 1;
    float* xyz_nxt  = (i & 1) ? xyzB  : xyzA;
    float* feat_nxt = (i & 1) ? featA : featB;

    zero_kernel<<<1, 64, 0, stream>>>(stats, 64);
    zero_kernel<<<(2 * Cout + 255) / 256, 256, 0, stream>>>(bnst, 2 * Cout);

    fps_kernel<<<16, 256, 0, stream>>>(xyz_cur, N, S, fpsidx, xyz_nxt);

    knn_kernel<<<dim3(S / 16, 16), 32, 0, stream>>>(xyz_cur, xyz_nxt, N, S, knnidx);

    center_stats_kernel<<<512, 256, 0, stream>>>(xyz_cur, xyz_nxt, feat_cur,
                                                 fpsidx, knnidx, N, S, C, stats);

    stage_main_kernel<<<dim3(S, 16), 256, 0, stream>>>(
        xyz_cur, xyz_nxt, feat_cur, fpsidx, knnidx, stats,
        N, S, C, Cout, Fs[i], FNs[i], feat_nxt, bnst);

    {
      int total = 16 * S * Cout;
      bn_gelu_kernel<<<(total + 255) / 256, 256, 0, stream>>>(
          feat_nxt, bnst, gammas[i], betas[i], S, Cout);
    }

    xyz_cur = xyz_nxt;
    feat_cur = feat_nxt;
    N = S; C = Cout;
  }

  final_kernel<<<dim3(4, 16), 256, 0, stream>>>(feat_cur, (float*)d_out);
}